// CustomLoss_49057116455661
// MI455X (gfx1250) — compile-verified
//
#include <hip/hip_runtime.h>

#define HH 512
#define WW 512
#define NPIX (HH * WW)
#define BIG_I (1 << 30)
#define BIG_F 1.0e6f

typedef float v2f __attribute__((ext_vector_type(2)));
typedef float v8f __attribute__((ext_vector_type(8)));

// ---- workspace layout (bytes) ----
// [0, 1MB)              : parent (int[NPIX])     union-find / labels
// [1MB, 2MB)            : dt     (float[NPIX])   distance transform (in-place)
// [2MB, +64)            : Scal
// [+64, +64+1KB)        : wavePartial (float[256])   soa partial sums
// next 2KB              : colMin (float[512])
// next 2KB              : colSum (float[512])
struct Scal {
  int sl;
  int el;
  int both;
  float r0;
  float r1;
};

__device__ __forceinline__ bool fg_of(float v) { return rintf(v) > 0.5f; }

// ---------------- init: parent + scalar init ----------------
__global__ void k_init(const float* __restrict__ img, int* __restrict__ parent, Scal* S) {
  int i = blockIdx.x * blockDim.x + threadIdx.x;
  if (i < NPIX) parent[i] = fg_of(img[i]) ? i : BIG_I;
  if (i == 0) { S->sl = -1; S->el = -1; S->both = 0; S->r0 = 0.f; S->r1 = 0.f; }
}

// ---------------- soa = sum(1 - result) via WMMA f32 16x16x4 ----------------
// 16 blocks x 256 threads = 128 waves, 2048 elems/wave. B = ones => D[m][n] = row-sum of A.
// C/D layout: VGPR v holds M=v (lanes 0-15) and M=v+8 (lanes 16-31); columns identical.
__global__ void k_soa(const float* __restrict__ img, float* __restrict__ wavePartial) {
  int gtid = blockIdx.x * blockDim.x + threadIdx.x;
  int wave = gtid >> 5;
  int lane = threadIdx.x & 31;
  int base = wave * (NPIX / 128);
#if defined(__HIP_DEVICE_COMPILE__) && __has_builtin(__builtin_amdgcn_wmma_f32_16x16x4_f32)
  v8f acc = {0.f, 0.f, 0.f, 0.f, 0.f, 0.f, 0.f, 0.f};
  v2f b;
  b[0] = 1.0f; b[1] = 1.0f;
  for (int it = 0; it < (NPIX / 128) / 64; ++it) {  // 32 iterations, 64 elems each
    int off = base + it * 64 + lane;
    v2f a;
    a[0] = 1.0f - img[off];
    a[1] = 1.0f - img[off + 32];
    acc = __builtin_amdgcn_wmma_f32_16x16x4_f32(false, a, false, b, (short)0, acc,
                                                false, false);
  }
  float s = acc[0] + acc[1] + acc[2] + acc[3] + acc[4] + acc[5] + acc[6] + acc[7];
  // lane 0 holds sum over M=0..7 (col 0), lane 16 holds sum over M=8..15
  if ((lane & 15) == 0) wavePartial[2 * wave + (lane >> 4)] = s;
#else
  float s = 0.f;
  for (int k = lane; k < NPIX / 128; k += 32) s += 1.0f - img[base + k];
  for (int o = 16; o > 0; o >>= 1) s += __shfl_down(s, o, 32);
  if (lane == 0) { wavePartial[2 * wave] = s; wavePartial[2 * wave + 1] = 0.f; }
#endif
}

// ---------------- union-find CCL (exact, fixed pass count) ----------------
__device__ __forceinline__ int rep(int* p, int x) {
  volatile int* vp = (volatile int*)p;
  int r = x;
  int pr = vp[r];
  while (pr != r) { r = pr; pr = vp[r]; }
  return r;
}

__device__ __forceinline__ void unite(int* p, int a, int b) {
  int ra = rep(p, a), rb = rep(p, b);
  while (ra != rb) {
    int hi = ra > rb ? ra : rb;
    int lo = ra > rb ? rb : ra;
    int old = atomicMin(&p[hi], lo);
    if (old == hi) return;  // attached hi under lo
    ra = lo; rb = old;      // keep merging the displaced link
  }
}

__global__ void k_merge(const float* __restrict__ img, int* parent) {
  int i = blockIdx.x * blockDim.x + threadIdx.x;
  if (i >= NPIX) return;
  if (!fg_of(img[i])) return;
  int r = i / WW, c = i % WW;
  if (c + 1 < WW && fg_of(img[i + 1])) unite(parent, i, i + 1);            // E
  if (r + 1 < HH) {
    if (c > 0 && fg_of(img[i + WW - 1])) unite(parent, i, i + WW - 1);     // SW
    if (fg_of(img[i + WW])) unite(parent, i, i + WW);                      // S
    if (c + 1 < WW && fg_of(img[i + WW + 1])) unite(parent, i, i + WW + 1);// SE
  }
}

__global__ void k_flatten(int* parent) {
  int i = blockIdx.x * blockDim.x + threadIdx.x;
  if (i >= NPIX) return;
  if (parent[i] != BIG_I) parent[i] = rep(parent, i);  // root == min index of component
}

// ---------------- fetch point labels ----------------
__global__ void k_points(const float* __restrict__ img, const int* __restrict__ pts,
                         const int* __restrict__ parent, Scal* S) {
  if (threadIdx.x != 0 || blockIdx.x != 0) return;
  // batch 3 of points_given (4,2,2) -> flat offset 12
  int p0 = pts[12] * WW + pts[13];
  int p1 = pts[14] * WW + pts[15];
  float r0 = img[p0], r1 = img[p1];
  bool f0 = fg_of(r0), f1 = fg_of(r1);
  S->r0 = r0;
  S->r1 = r1;
  S->sl = f0 ? parent[p0] : -1;  // -1 never matches a parent value
  S->el = f1 ? parent[p1] : -1;
  S->both = (f0 && f1) ? 1 : 0;
}

// ---------------- exact separable L1 DT (4 in-place scans) ----------------
__global__ void k_dt_row_fwd(const int* __restrict__ parent, float* __restrict__ dt,
                             const Scal* S) {
  int r = blockIdx.x * blockDim.x + threadIdx.x;
  if (r >= HH) return;
  int el = S->el;
  float run = BIG_F;
  for (int c = 0; c < WW; ++c) {
    int idx = r * WW + c;
    float d0 = (parent[idx] == el) ? 0.f : BIG_F;
    run = fminf(d0, run + 1.f);
    dt[idx] = run;
  }
}

__global__ void k_dt_row_bwd(float* __restrict__ dt) {
  int r = blockIdx.x * blockDim.x + threadIdx.x;
  if (r >= HH) return;
  float run = BIG_F;
  for (int c = WW - 1; c >= 0; --c) {
    int idx = r * WW + c;
    run = fminf(dt[idx], run + 1.f);
    dt[idx] = run;
  }
}

__global__ void k_dt_col_fwd(float* __restrict__ dt) {
  int c = blockIdx.x * blockDim.x + threadIdx.x;
  if (c >= WW) return;
  float run = BIG_F;
  for (int r = 0; r < HH; ++r) {
    int idx = r * WW + c;
    run = fminf(dt[idx], run + 1.f);
    dt[idx] = run;
  }
}

// final column pass; also per-column start-cluster min/sum (deterministic slots)
__global__ void k_dt_col_bwd(const float* __restrict__ img, const int* __restrict__ parent,
                             float* __restrict__ dt, const Scal* S,
                             float* __restrict__ colMin, float* __restrict__ colSum) {
  int c = blockIdx.x * blockDim.x + threadIdx.x;
  if (c >= WW) return;
  int sl = S->sl;
  float run = BIG_F;
  float cmin = 4.f * BIG_F;
  float csum = 0.f;
  for (int r = HH - 1; r >= 0; --r) {
    int idx = r * WW + c;
    run = fminf(dt[idx], run + 1.f);
    dt[idx] = run;
    if (parent[idx] == sl) {  // start_mask (implies fg; sl==-1 never matches)
      cmin = fminf(cmin, run);
      csum += img[idx];
    }
  }
  colMin[c] = cmin;
  colSum[c] = csum;
}

// ---------------- finalize (single thread, fixed-order sums) ----------------
__global__ void k_final(const float* __restrict__ dt, const int* __restrict__ pts,
                        const Scal* __restrict__ S, const float* __restrict__ wavePartial,
                        const float* __restrict__ colMin, const float* __restrict__ colSum,
                        float* __restrict__ out) {
  if (threadIdx.x != 0 || blockIdx.x != 0) return;
  float soa = 0.f;
  for (int k = 0; k < 256; ++k) soa += wavePartial[k];
  float cmin = 4.f * BIG_F, csum = 0.f;
  for (int c = 0; c < WW; ++c) { cmin = fminf(cmin, colMin[c]); csum += colSum[c]; }
  int p0 = pts[12] * WW + pts[13];
  float min_d = fminf(cmin, dt[p0]);
  float fallback = (2.0f - (S->r0 + S->r1)) * 100.0f;
  float gap = S->both ? (min_d * soa * 10.0f * soa) : fallback;
  float clp = S->both ? (csum * 90.0f) : fallback;
  out[0] = fallback;  // loss_start
  out[1] = gap;       // gap_loss
  out[2] = clp;       // cluster_size_penalty
}

extern "C" void kernel_launch(void* const* d_in, const int* in_sizes, int n_in,
                              void* d_out, int out_size, void* d_ws, size_t ws_size,
                              hipStream_t stream) {
  const float* result = (const float*)d_in[0];   // (4,1,512,512) f32
  const int* pts = (const int*)d_in[1];          // (4,2,2) i32
  const float* img = result + (size_t)3 * NPIX;  // only last batch matters
  float* out = (float*)d_out;

  char* ws = (char*)d_ws;
  int* parent = (int*)ws;
  float* dt = (float*)(ws + (size_t)NPIX * 4);
  Scal* S = (Scal*)(ws + (size_t)NPIX * 8);
  float* wavePartial = (float*)(ws + (size_t)NPIX * 8 + 64);
  float* colMin = (float*)(ws + (size_t)NPIX * 8 + 64 + 256 * 4);
  float* colSum = (float*)(ws + (size_t)NPIX * 8 + 64 + 256 * 4 + 512 * 4);

  dim3 b256(256);
  dim3 gPix((NPIX + 255) / 256);
  dim3 gLine(2), bLine(256);

  k_init<<<gPix, b256, 0, stream>>>(img, parent, S);
  k_soa<<<dim3(16), b256, 0, stream>>>(img, wavePartial);
  k_merge<<<gPix, b256, 0, stream>>>(img, parent);
  k_flatten<<<gPix, b256, 0, stream>>>(parent);
  k_points<<<dim3(1), dim3(32), 0, stream>>>(img, pts, parent, S);
  k_dt_row_fwd<<<gLine, bLine, 0, stream>>>(parent, dt, S);
  k_dt_row_bwd<<<gLine, bLine, 0, stream>>>(dt);
  k_dt_col_fwd<<<gLine, bLine, 0, stream>>>(dt);
  k_dt_col_bwd<<<gLine, bLine, 0, stream>>>(img, parent, dt, S, colMin, colSum);
  k_final<<<dim3(1), dim3(32), 0, stream>>>(dt, pts, S, wavePartial, colMin, colSum, out);
}